// EfficientMemoryGELU_42073499632317
// MI455X (gfx1250) — compile-verified
//
#include <hip/hip_runtime.h>

// Native clang vector type: accepted by __builtin_nontemporal_* and lowers to
// 128-bit global memory ops (b128) with th:NT cache hints.
typedef float v4f __attribute__((ext_vector_type(4)));

// Branchless exact-erf GELU.
// Abramowitz-Stegun 7.1.26: erf(z) = 1 - P5(t)*exp(-z^2), t = 1/(1+p*z), z>=0,
// |abs error| <= 1.5e-7 -> GELU abs error <~ 1e-7*|x|, i.e. fp32-ulp level vs
// jax.nn.gelu(approximate=False). No data-dependent branches, so no exec-mask
// divergence (the OCML erff is piecewise and executes both paths per wave).
__device__ __forceinline__ float gelu_exact(float x) {
    const float z  = fabsf(x) * 0.70710678118654752440f;   // |x|/sqrt(2)
    const float t  = __builtin_amdgcn_rcpf(fmaf(0.3275911f, z, 1.0f)); // v_rcp_f32
    float p = fmaf(1.061405429f, t, -1.453152027f);
    p = fmaf(p, t, 1.421413741f);
    p = fmaf(p, t, -0.284496736f);
    p = fmaf(p, t, 0.254829592f);
    p *= t;
    const float e       = __expf(-z * z);                  // v_exp_f32 path
    const float erf_abs = fmaf(-p, e, 1.0f);               // erf(|z|)
    const float erf_z   = copysignf(erf_abs, x);           // v_bfi_b32
    return 0.5f * x * (1.0f + erf_z);
}

// Streaming elementwise GELU, fp32 -> fp32.
// Memory-bound: 536MB of traffic vs 192MB L2 and zero reuse -> use 128-bit
// non-temporal loads/stores so the stream doesn't thrash the cache hierarchy.
__global__ __launch_bounds__(256) void gelu_f32_nt_kernel(
    const float* __restrict__ x, float* __restrict__ y,
    long long n4, long long n) {
    const long long idx    = (long long)blockIdx.x * blockDim.x + threadIdx.x;
    const long long stride = (long long)gridDim.x * blockDim.x;

    const v4f* __restrict__ xv = (const v4f*)x;
    v4f* __restrict__ yv       = (v4f*)y;

    // Main vectorized grid-stride loop: one global_load_b128 (th:NT),
    // branchless erf VALU body, one global_store_b128 (th:NT) per iteration.
    for (long long i = idx; i < n4; i += stride) {
        v4f v = __builtin_nontemporal_load(xv + i);
        v4f r;
        r.x = gelu_exact(v.x);
        r.y = gelu_exact(v.y);
        r.z = gelu_exact(v.z);
        r.w = gelu_exact(v.w);
        __builtin_nontemporal_store(r, yv + i);
    }

    // Scalar tail for n not divisible by 4 (n is divisible here, but stay general).
    const long long rem_base = n4 * 4;
    const long long rem      = n - rem_base;
    if (idx < rem) {
        y[rem_base + idx] = gelu_exact(x[rem_base + idx]);
    }
}

extern "C" void kernel_launch(void* const* d_in, const int* in_sizes, int n_in,
                              void* d_out, int out_size, void* d_ws, size_t ws_size,
                              hipStream_t stream) {
    (void)n_in; (void)d_ws; (void)ws_size; (void)out_size;

    const float* x = (const float*)d_in[0];
    float*       y = (float*)d_out;

    const long long n  = (long long)in_sizes[0];   // 4*4096*4096 = 67,108,864
    const long long n4 = n / 4;                    // float4 count = 16,777,216

    const int threads = 256;                       // 8 wave32 per block
    long long want_blocks = (n4 + threads - 1) / threads;
    if (want_blocks < 1) want_blocks = 1;
    // Cap the grid; grid-stride loop covers the rest. 65536 blocks x 256 thr
    // = 16.7M threads -> exactly one float4 per thread at this problem size,
    // maximizing the number of independent B128 loads in flight.
    const int blocks = (int)(want_blocks > 65536 ? 65536 : want_blocks);

    gelu_f32_nt_kernel<<<blocks, threads, 0, stream>>>(x, y, n4, n);
}